// LNRNet_72318659330244
// MI455X (gfx1250) — compile-verified
//
#include <hip/hip_runtime.h>
#include <stdint.h>

#define B_    4
#define H_    240
#define W_    320
#define HW    76800
#define FE_   86
#define OUTC  89
#define NE_   3
#define ME_   4
#define LE_   10
#define PE_   16
#define SE_   20
#define CE_   22
#define F_    64
#define R_    32
#define R3    32768
#define THRESH 0.7f
#define BIGF  1e30f

typedef __attribute__((ext_vector_type(2))) float v2f;
typedef __attribute__((ext_vector_type(8))) float v8f;

__device__ __forceinline__ float sigm(float x){ return 1.0f/(1.0f + expf(-x)); }

// monotone float <-> ordered-uint mapping (for atomicMin/Max on floats)
__device__ __forceinline__ uint32_t fkey(float f){
  uint32_t u = __float_as_uint(f);
  return (u & 0x80000000u) ? ~u : (u | 0x80000000u);
}
__device__ __forceinline__ float funkey(uint32_t k){
  uint32_t u = (k & 0x80000000u) ? (k ^ 0x80000000u) : ~k;
  return __uint_as_float(u);
}

// ---- K0: zero occ grid, zero sum accumulators, init min/max keys ----
__global__ void k_init(float* __restrict__ occ, float* __restrict__ acc,
                       uint32_t* __restrict__ mmkey){
  size_t i = (size_t)blockIdx.x*blockDim.x + threadIdx.x;
  const size_t n = (size_t)B_*F_*R3;
  for (size_t x = i; x < n; x += (size_t)gridDim.x*blockDim.x) occ[x] = 0.0f;
  if (i < 56) acc[i] = 0.0f;
  if (i >= 64 && i < 88){
    int k = (int)(i - 64) % 6;
    mmkey[i - 64] = (k < 3) ? fkey(BIGF) : fkey(-BIGF);
  }
}

// ---- K1: per-(b,j) score sums + vote numerators (14 scalars per batch) ----
__global__ void k_sums(const float* __restrict__ in, float* __restrict__ acc){
  const int b = blockIdx.y;
  const float* ib = in + (size_t)b*FE_*HW;
  float ls[14];
  #pragma unroll
  for (int k=0;k<14;k++) ls[k]=0.0f;
  for (int p = blockIdx.x*blockDim.x + threadIdx.x; p < HW; p += gridDim.x*blockDim.x){
    float m  = sigm(ib[NE_*HW + p]);
    float m2 = m*m;
    #pragma unroll
    for (int j=0;j<2;j++){
      float sg = sigm(ib[(SE_+j)*HW + p]);
      ls[j] += sg*m;              // score_raw sum
      float w = sg*m2;            // jmap weight
      #pragma unroll
      for (int c=0;c<3;c++){
        ls[2 + j*3 + c] += ib[(ME_ + j*3 + c)*HW + p] * w;   // loc numerator
        ls[8 + j*3 + c] += ib[(LE_ + j*3 + c)*HW + p] * w;   // rot numerator
      }
    }
  }
  #pragma unroll
  for (int k=0;k<14;k++){
    float v = ls[k];
    for (int off=16; off>0; off>>=1) v += __shfl_down(v, off, 32);
    if ((threadIdx.x & 31) == 0) atomicAdd(&acc[b*14 + k], v);
  }
}

// ---- K2: Rodrigues -> [R|t] rows; Mst[b][mi*3+p][q], 12 matrices total ----
__global__ void k_mats(const float* __restrict__ acc, float* __restrict__ Mst){
  int t = threadIdx.x;
  if (t >= B_*3) return;
  int b = t/3, mi = t%3;
  float loc[3] = {0,0,0}, rot[3] = {0,0,0};
  if (mi > 0){
    int j = mi - 1;
    float S = acc[b*14 + j] + 1e-5f;
    #pragma unroll
    for (int c=0;c<3;c++){
      loc[c] = acc[b*14 + 2 + j*3 + c] / S;
      rot[c] = acc[b*14 + 8 + j*3 + c] / S;
    }
  }
  float r0=-rot[0], r1=-rot[1], r2=-rot[2];
  float ang = sqrtf(r0*r0 + r1*r1 + r2*r2) + 1e-8f;
  float x = r0/ang, y = r1/ang, z = r2/ang;
  float c = cosf(ang), s = sinf(ang), oc = 1.0f - c;
  float Rm[3][3];
  Rm[0][0] = 1.0f + oc*(-(y*y+z*z));
  Rm[0][1] = s*(-z) + oc*(x*y);
  Rm[0][2] = s*( y) + oc*(x*z);
  Rm[1][0] = s*( z) + oc*(x*y);
  Rm[1][1] = 1.0f + oc*(-(x*x+z*z));
  Rm[1][2] = s*(-x) + oc*(y*z);
  Rm[2][0] = s*(-y) + oc*(x*z);
  Rm[2][1] = s*( x) + oc*(y*z);
  Rm[2][2] = 1.0f + oc*(-(x*x+y*y));
  float tr[3];
  #pragma unroll
  for (int p=0;p<3;p++)
    tr[p] = loc[p] - (Rm[p][0]*loc[0] + Rm[p][1]*loc[1] + Rm[p][2]*loc[2]);
  #pragma unroll
  for (int p=0;p<3;p++){
    float* row = &Mst[(size_t)(b*9 + mi*3 + p)*4];
    row[0]=Rm[p][0]; row[1]=Rm[p][1]; row[2]=Rm[p][2]; row[3]=tr[p];
  }
}

// ---- K3: pn via V_WMMA_F32_16X16X4_F32 ----
// A (16x4): rows 0..8 = the 9 matrix rows [R|t] (loop-invariant), rows 9..15 = 0.
// B (4x16): 16 pixels' homog vectors [n0,n1,n2,1] per issue.
// D (16x16): lane L<16 holds rows 0..7 for pixel (base+L); row 8 comes from lane L+16.
__global__ void k_pn(const float* __restrict__ in, const float* __restrict__ Mst,
                     float* __restrict__ pnbuf, uint32_t* __restrict__ mmkey){
  const int b = blockIdx.y;
  const float* ib = in + (size_t)b*FE_*HW;
  const int lane = threadIdx.x & 31;
  const int wave = threadIdx.x >> 5;
  const int wgid = blockIdx.x*8 + wave;     // 0..479 per batch
  const int row  = lane & 15;
  const int hi   = lane >> 4;               // K-half select
  v2f a;
  if (row < 9){
    const float* mr = &Mst[(size_t)(b*9 + row)*4 + hi*2];
    a.x = mr[0]; a.y = mr[1];
  } else { a.x = 0.0f; a.y = 0.0f; }
  float lo0=BIGF, lo1=BIGF, lo2=BIGF, up0=-BIGF, up1=-BIGF, up2=-BIGF;
  for (int it=0; it<10; ++it){              // 480 waves * 10 * 16 px = 76800 exactly
    int p = (wgid + it*480)*16 + row;
    v2f bb;
    if (hi == 0){ bb.x = ib[0*HW + p]; bb.y = ib[1*HW + p]; }
    else        { bb.x = ib[2*HW + p]; bb.y = 1.0f; }
    v8f cacc = {};
    v8f d = __builtin_amdgcn_wmma_f32_16x16x4_f32(
        false, a, false, bb, (short)0, cacc, false, false);
    float row8 = __shfl_xor(d[0], 16, 32);  // candidate (mi=2,p=2) for my pixel
    if (hi == 0){
      float s0 = ib[(PE_+0)*HW + p];
      float s1 = ib[(PE_+1)*HW + p];
      float s2 = ib[(PE_+2)*HW + p];
      float s3 = ib[(PE_+3)*HW + p];
      int am = 0; float bst = s0;
      if (s1 > bst){ bst = s1; am = 1; }
      if (s2 > bst){ bst = s2; am = 2; }
      if (s3 > bst){ bst = s3; am = 3; }
      float pn0=0.0f, pn1=0.0f, pn2=0.0f;
      if      (am == 1){ pn0 = d[0]; pn1 = d[1]; pn2 = d[2]; }
      else if (am == 2){ pn0 = d[3]; pn1 = d[4]; pn2 = d[5]; }
      else if (am == 3){ pn0 = d[6]; pn1 = d[7]; pn2 = row8; }
      pnbuf[(size_t)(b*3+0)*HW + p] = pn0;
      pnbuf[(size_t)(b*3+1)*HW + p] = pn1;
      pnbuf[(size_t)(b*3+2)*HW + p] = pn2;
      float m = sigm(ib[NE_*HW + p]);
      if (m > THRESH){
        lo0 = fminf(lo0, pn0); lo1 = fminf(lo1, pn1); lo2 = fminf(lo2, pn2);
        up0 = fmaxf(up0, pn0); up1 = fmaxf(up1, pn1); up2 = fmaxf(up2, pn2);
      }
    }
  }
  if (hi == 0){
    atomicMin(&mmkey[b*6+0], fkey(lo0));
    atomicMin(&mmkey[b*6+1], fkey(lo1));
    atomicMin(&mmkey[b*6+2], fkey(lo2));
    atomicMax(&mmkey[b*6+3], fkey(up0));
    atomicMax(&mmkey[b*6+4], fkey(up1));
    atomicMax(&mmkey[b*6+5], fkey(up2));
  }
}

// ---- K4: per-batch low/scale/safe ----
__global__ void k_params(const uint32_t* __restrict__ mmkey, float* __restrict__ par){
  int b = threadIdx.x;
  if (b >= B_) return;
  float lo[3], up[3];
  #pragma unroll
  for (int c=0;c<3;c++){ lo[c] = funkey(mmkey[b*6+c]); up[c] = funkey(mmkey[b*6+3+c]); }
  float scale = fmaxf(fmaxf(up[0]-lo[0], up[1]-lo[1]), up[2]-lo[2]);
  float safe  = (scale == 0.0f) ? 1.0f : scale;
  par[b*8+0]=lo[0]; par[b*8+1]=lo[1]; par[b*8+2]=lo[2];
  par[b*8+3]=scale; par[b*8+4]=safe;
}

// ---- K5: pn_map channels + voxel scatter-add of 64 feature channels ----
__global__ void k_out(const float* __restrict__ in, const float* __restrict__ pnbuf,
                      const float* __restrict__ par, float* __restrict__ outcat,
                      float* __restrict__ occ){
  const int b = blockIdx.y;
  const float* ib = in + (size_t)b*FE_*HW;
  const float lo0 = par[b*8+0], lo1 = par[b*8+1], lo2 = par[b*8+2];
  const float scale = par[b*8+3], safe = par[b*8+4];
  const bool donorm = (scale != 0.0f);
  float* ob   = outcat + (size_t)b*OUTC*HW;
  float* occb = occ    + (size_t)b*F_*R3;
  for (int p = blockIdx.x*blockDim.x + threadIdx.x; p < HW; p += gridDim.x*blockDim.x){
    float pn0 = pnbuf[(size_t)(b*3+0)*HW + p];
    float pn1 = pnbuf[(size_t)(b*3+1)*HW + p];
    float pn2 = pnbuf[(size_t)(b*3+2)*HW + p];
    float n0 = donorm ? (pn0 - lo0)/safe : pn0;
    float n1 = donorm ? (pn1 - lo1)/safe : pn1;
    float n2 = donorm ? (pn2 - lo2)/safe : pn2;
    float m  = sigm(ib[NE_*HW + p]);
    bool msk = m > THRESH;
    float q0 = msk ? n0 : 0.0f;
    float q1 = msk ? n1 : 0.0f;
    float q2 = msk ? n2 : 0.0f;
    ob[(size_t)(FE_+0)*HW + p] = q0;
    ob[(size_t)(FE_+1)*HW + p] = q1;
    ob[(size_t)(FE_+2)*HW + p] = q2;
    if (msk){
      int v0 = (int)fminf(fmaxf(floorf(q0*(float)R_), 0.0f), (float)(R_-1));
      int v1 = (int)fminf(fmaxf(floorf(q1*(float)R_), 0.0f), (float)(R_-1));
      int v2 = (int)fminf(fmaxf(floorf(q2*(float)R_), 0.0f), (float)(R_-1));
      int flat = (v0*R_ + v1)*R_ + v2;
      #pragma unroll 8
      for (int f=0; f<F_; f++){
        atomicAdd(&occb[(size_t)f*R3 + flat], ib[(size_t)(CE_+f)*HW + p]);
      }
    }
  }
}

extern "C" void kernel_launch(void* const* d_in, const int* in_sizes, int n_in,
                              void* d_out, int out_size, void* d_ws, size_t ws_size,
                              hipStream_t stream){
  const float* in = (const float*)d_in[0];
  float* out = (float*)d_out;
  float* occ = out + (size_t)B_*OUTC*HW;

  float*    wsf   = (float*)d_ws;
  uint32_t* wsu   = (uint32_t*)d_ws;
  float*    acc   = wsf;            // [0,56)   per-batch sums
  uint32_t* mmkey = wsu + 64;       // [64,88)  min/max ordered keys
  float*    Mst   = wsf + 96;       // [96,240) 36 matrix rows x4
  float*    par   = wsf + 240;      // [240,272) low/scale/safe
  float*    pnbuf = wsf + 512;      // 4*3*76800 floats

  // verbatim copy of the 86 input channels into out_cat channels [0,86)
  for (int b=0; b<B_; b++){
    hipMemcpyAsync(out + (size_t)b*OUTC*HW, in + (size_t)b*FE_*HW,
                   (size_t)FE_*HW*sizeof(float), hipMemcpyDeviceToDevice, stream);
  }
  k_init  <<<512, 256, 0, stream>>>(occ, acc, mmkey);
  k_sums  <<<dim3(120, B_), 256, 0, stream>>>(in, acc);
  k_mats  <<<1, 16, 0, stream>>>(acc, Mst);
  k_pn    <<<dim3(60, B_), 256, 0, stream>>>(in, Mst, pnbuf, mmkey);
  k_params<<<1, B_, 0, stream>>>(mmkey, par);
  k_out   <<<dim3(120, B_), 256, 0, stream>>>(in, pnbuf, par, out, occ);
}